// GaussianAffinity_59253368816316
// MI455X (gfx1250) — compile-verified
//
#include <hip/hip_runtime.h>
#include <stdint.h>

typedef __attribute__((ext_vector_type(2))) float v2f;
typedef __attribute__((ext_vector_type(8))) float v8f;

// D = A(16x4 f32) * B(4x16 f32) + C(16x16 f32)
#define WMMA_F32X4(a, b, c) \
  __builtin_amdgcn_wmma_f32_16x16x4_f32(false, (a), false, (b), (short)0, (c), false, false)

// Problem shape (fixed by the reference)
constexpr int Bsz = 8;
constexpr int Dd  = 512;
constexpr int Nn  = 2048;
constexpr int Mm  = 2048;

// Tiling
constexpr int TN = 64;    // block tile rows (n)
constexpr int TM = 128;   // block tile cols (m)
constexpr int KC = 16;    // K chunk staged in LDS

// Async DMA of 16 bytes: global -> LDS, tracked by ASYNCcnt (CDNA5 §08).
// LDS byte address = low 32 bits of the generic shared pointer (flat->LDS map).
__device__ __forceinline__ void async_cp16(const float* g, float* lds) {
  const uint32_t l = (uint32_t)(uintptr_t)lds;
  asm volatile("global_load_async_to_lds_b128 %0, %1, off"
               :: "v"(l), "v"(g)
               : "memory");
}

__device__ __forceinline__ void wait_async_all() {
  asm volatile("s_wait_asynccnt 0x0" ::: "memory");
}

__global__ __launch_bounds__(256)
void gaussian_affinity_wmma(const float* __restrict__ X,
                            const float* __restrict__ Y,
                            const float* __restrict__ Ux,
                            const float* __restrict__ Uy,
                            float* __restrict__ out)
{
  // Double-buffered LDS staging (rows padded; 68*4 / 132*4 bytes are 16B multiples)
  __shared__ float Xs [2][KC][68];
  __shared__ float Uxs[2][KC][68];
  __shared__ float Ys [2][KC][132];
  __shared__ float Uys[2][KC][132];
  __shared__ float x2s[TN];
  __shared__ float y2s[TM];

  const int t    = threadIdx.x;
  const int lane = t & 31;
  const int wave = t >> 5;

  const int b  = blockIdx.z;
  const int bN = blockIdx.y * TN;
  const int bM = blockIdx.x * TM;

  // wave -> 2x4 arrangement of 32x32 patches inside the 64x128 block tile
  const int wn = (wave >> 2) * 32;  // 0 or 32
  const int wm = (wave & 3)  * 32;  // 0,32,64,96

  v8f accE[2][2] = {};  // Xt*Y accumulators
  v8f accP[2][2] = {};  // Uxt*Uy accumulators

  // staging map (float4 granularity)
  const int xr = t >> 4;         // 0..15 : K row for X/Ux
  const int xc = (t & 15) * 4;   // 0..60 : col4 for X/Ux (16 float4 per row)
  const int yr0 = t >> 5;        // Y/Uy slot 0: rows 0..7
  const int yc0 = (t & 31) * 4;  //               cols 0..124
  const int yr1 = yr0 + 8;       // Y/Uy slot 1: rows 8..15

  float sx = 0.0f, sy = 0.0f;    // per-thread partial squared norms

  const size_t strideB = (size_t)Dd * Nn;   // D*N == D*M
  const float* Xb  = X  + (size_t)b * strideB;
  const float* Yb  = Y  + (size_t)b * strideB;
  const float* Uxb = Ux + (size_t)b * strideB;
  const float* Uyb = Uy + (size_t)b * strideB;

  const int halfk = (lane >> 4) * 2;  // lanes 0-15 -> K0/K1, lanes 16-31 -> K2/K3
  const int l16   = lane & 15;

  // Issue the 6 per-thread async DMAs for K-chunk k0 into buffer s.
  auto stage = [&](int k0, int s) {
    async_cp16(Xb  + (size_t)(k0 + xr)  * Nn + bN + xc,  &Xs [s][xr][xc]);
    async_cp16(Uxb + (size_t)(k0 + xr)  * Nn + bN + xc,  &Uxs[s][xr][xc]);
    async_cp16(Yb  + (size_t)(k0 + yr0) * Mm + bM + yc0, &Ys [s][yr0][yc0]);
    async_cp16(Uyb + (size_t)(k0 + yr0) * Mm + bM + yc0, &Uys[s][yr0][yc0]);
    async_cp16(Yb  + (size_t)(k0 + yr1) * Mm + bM + yc0, &Ys [s][yr1][yc0]);
    async_cp16(Uyb + (size_t)(k0 + yr1) * Mm + bM + yc0, &Uys[s][yr1][yc0]);
  };

  constexpr int NCHUNK = Dd / KC;
  stage(0, 0);  // prologue: chunk 0 in flight

  for (int c = 0; c < NCHUNK; ++c) {
    const int cur = c & 1;

    // Wait for this wave's in-flight DMA (only chunk c outstanding), then make
    // all waves' LDS writes visible. The barrier also proves every wave is done
    // reading buffer cur^1 (its compute of iter c-1 precedes this barrier), so
    // issuing DMA into cur^1 below is race-free and overlaps with our compute.
    wait_async_all();
    __syncthreads();
    if (c + 1 < NCHUNK) stage((c + 1) * KC, cur ^ 1);

    // ---------------- squared-norm partials (reuse staged tiles) ----------------
    if (t < TN) {                      // waves 0-1: x2 over the 64 n's
#pragma unroll
      for (int k = 0; k < KC; ++k) { const float v = Xs[cur][k][t]; sx += v * v; }
    } else if (t < TN + TM) {          // waves 2-5: y2 over the 128 m's
      const int m = t - TN;
#pragma unroll
      for (int k = 0; k < KC; ++k) { const float v = Ys[cur][k][m]; sy += v * v; }
    }

    // ---------------- WMMA: 4 k-steps x (2x2 tiles) x 2 GEMMs --------------------
#pragma unroll
    for (int kk = 0; kk < KC; kk += 4) {
      const int ka = kk + halfk;
      v2f aE[2], aP[2], bE[2], bP[2];
#pragma unroll
      for (int i = 0; i < 2; ++i) {
        const int n = wn + i * 16 + l16;
        aE[i].x = Xs [cur][ka][n];  aE[i].y = Xs [cur][ka + 1][n];
        aP[i].x = Uxs[cur][ka][n];  aP[i].y = Uxs[cur][ka + 1][n];
        const int m = wm + i * 16 + l16;
        bE[i].x = Ys [cur][ka][m];  bE[i].y = Ys [cur][ka + 1][m];
        bP[i].x = Uys[cur][ka][m];  bP[i].y = Uys[cur][ka + 1][m];
      }
#pragma unroll
      for (int i = 0; i < 2; ++i)
#pragma unroll
        for (int j = 0; j < 2; ++j) {
          accE[i][j] = WMMA_F32X4(aE[i], bE[j], accE[i][j]);
          accP[i][j] = WMMA_F32X4(aP[i], bP[j], accP[i][j]);
        }
    }
    // no trailing barrier: next iteration's wait+barrier covers the WAR hazard
  }

  // publish squared norms
  if (t < TN)            x2s[t]       = sx;
  else if (t < TN + TM)  y2s[t - TN]  = sy;
  __syncthreads();

  // ---------------- fused epilogue: Me = exp(-dist), Mp = accP --------------------
  // Streaming write-once outputs -> non-temporal stores (keep L2 for input reuse).
  const int  halfr  = (lane >> 4) * 8;               // C/D layout: lanes 16-31 hold M=r+8
  const size_t MeBase = (size_t)b * Nn * Mm;
  const size_t MpBase = (size_t)Bsz * Nn * Mm + MeBase;

#pragma unroll
  for (int i = 0; i < 2; ++i) {
#pragma unroll
    for (int j = 0; j < 2; ++j) {
      const int mloc = wm + j * 16 + l16;
      const int m    = bM + mloc;
      const float y2 = y2s[mloc];
#pragma unroll
      for (int r = 0; r < 8; ++r) {
        const int nloc = wn + i * 16 + r + halfr;
        const int n    = bN + nloc;
        const float xy = accE[i][j][r];
        float dist = x2s[nloc] + y2 - 2.0f * xy;
        dist = __builtin_isnan(dist) ? __builtin_inff() : dist;      // NaN -> inf mask
        __builtin_nontemporal_store(__expf(-dist),
                                    &out[MeBase + (size_t)n * Mm + m]); // SIGMA=AE=1
        __builtin_nontemporal_store(accP[i][j][r],
                                    &out[MpBase + (size_t)n * Mm + m]); // AP=1
      }
    }
  }
}

extern "C" void kernel_launch(void* const* d_in, const int* in_sizes, int n_in,
                              void* d_out, int out_size, void* d_ws, size_t ws_size,
                              hipStream_t stream) {
  (void)in_sizes; (void)n_in; (void)out_size; (void)d_ws; (void)ws_size;
  const float* X  = (const float*)d_in[0];
  const float* Y  = (const float*)d_in[1];
  const float* Ux = (const float*)d_in[2];
  const float* Uy = (const float*)d_in[3];
  float* out = (float*)d_out;

  dim3 grid(Mm / TM, Nn / TN, Bsz);   // (16, 32, 8)
  dim3 block(256);                    // 8 waves (wave32)
  gaussian_affinity_wmma<<<grid, block, 0, stream>>>(X, Y, Ux, Uy, out);
}